// DQC1_Model_17265768530267
// MI455X (gfx1250) — compile-verified
//
#include <hip/hip_runtime.h>
#include <hip/hip_bf16.h>

typedef float v2f __attribute__((ext_vector_type(2)));
typedef float v8f __attribute__((ext_vector_type(8)));

// ---------------- small complex helpers (setup kernel only) ----------------
struct cpx { float re, im; };
__device__ inline cpx cmul(cpx a, cpx b){ return {a.re*b.re - a.im*b.im, a.re*b.im + a.im*b.re}; }
__device__ inline cpx cadd(cpx a, cpx b){ return {a.re+b.re, a.im+b.im}; }

__device__ inline void mm2(const cpx A[2][2], const cpx B[2][2], cpx C[2][2]){
  for(int i=0;i<2;++i) for(int j=0;j<2;++j)
    C[i][j] = cadd(cmul(A[i][0],B[0][j]), cmul(A[i][1],B[1][j]));
}

// RZ(p2) @ RY(p1) @ RX(p0)
__device__ inline void euler_u2(float p0, float p1, float p2, cpx U[2][2]){
  float cx = __cosf(0.5f*p0), sx = __sinf(0.5f*p0);
  float cy = __cosf(0.5f*p1), sy = __sinf(0.5f*p1);
  float cz = __cosf(0.5f*p2), sz = __sinf(0.5f*p2);
  cpx RX[2][2] = {{{cx,0.f},{0.f,-sx}},{{0.f,-sx},{cx,0.f}}};
  cpx RY[2][2] = {{{cy,0.f},{-sy,0.f}},{{sy,0.f},{cy,0.f}}};
  cpx RZ[2][2] = {{{cz,-sz},{0.f,0.f}},{{0.f,0.f},{cz,sz}}};
  cpx T[2][2];
  mm2(RY, RX, T);
  mm2(RZ, T, U);
}

// Build per-layer WMMA A-operand tables (row-permuted realified W) into d_ws.
// Layout per layer k (128 floats): part p in 0..3 is the p-th A VGPR,
// lane m (<16) = A16[m][clo[p]], lane m+16 = A16[m][chi[p]].
__global__ void dqc1_build_A(const float* __restrict__ phi,
                             float* __restrict__ Atab, int nlayers){
  int k = threadIdx.x;
  if (k >= nlayers) return;
  const float* p = phi + 6*k;
  cpx A2[2][2], B2[2][2];
  euler_u2(p[0], p[1], p[2], A2);   // wire 1
  euler_u2(p[3], p[4], p[5], B2);   // wire 2
  cpx W[4][4];
  for(int i1=0;i1<2;++i1) for(int i2=0;i2<2;++i2)
    for(int j1=0;j1<2;++j1) for(int j2=0;j2<2;++j2)
      W[2*i1+i2][2*j1+j2] = cmul(A2[i1][j1], B2[i2][j2]);
  for(int j=0;j<4;++j){ W[3][j].re = -W[3][j].re; W[3][j].im = -W[3][j].im; } // CZ

  // M8 = [[Wr, -Wi],[Wi, Wr]] acting on s = [Re v0..3, Im v0..3]
  float M8[8][8];
  for(int i=0;i<4;++i) for(int j=0;j<4;++j){
    M8[i][j]     = W[i][j].re;  M8[i][j+4]   = -W[i][j].im;
    M8[i+4][j]   = W[i][j].im;  M8[i+4][j+4] =  W[i][j].re;
  }
  // Row-permuted 16x8 A so D vgprs 0..3 land directly in B-layout:
  // dest rows [0,1,2,3] = M8 rows [0,1,4,5]; dest rows [8..11] = M8 rows [2,3,6,7]
  float A16[16][8];
  for(int i=0;i<16;++i) for(int j=0;j<8;++j) A16[i][j] = 0.f;
  const int rmap[8] = {0,1,4,5, 2,3,6,7};
  for(int r=0;r<4;++r) for(int j=0;j<8;++j){
    A16[r][j]   = M8[rmap[r]][j];
    A16[r+8][j] = M8[rmap[4+r]][j];
  }
  const int clo[4] = {0,1,4,5}, chi[4] = {2,3,6,7};
  float* dst = Atab + k*128;
  for(int part=0; part<4; ++part)
    for(int m=0;m<16;++m){
      dst[part*32 + m]      = A16[m][clo[part]];
      dst[part*32 + 16 + m] = A16[m][chi[part]];
    }
}

// ---------------- main kernel: one wave = 16 data points ----------------
__global__ void __launch_bounds__(256)
dqc1_main(const float* __restrict__ x, const float* __restrict__ theta,
          const float* __restrict__ Atab, float* __restrict__ out, int n){
  const int lane = (int)(threadIdx.x & 31u);
  const int wave = (int)((blockIdx.x * blockDim.x + threadIdx.x) >> 5);
  const int base = wave * 16;
  if (base >= n) return;                       // whole-wave exit only

  const float xv = x[base + (lane & 15)];      // point value in both lane-halves

  // State in B-layout: b0={Re v0|Re v2}, b1={Re v1|Re v3}, b2={Im v0|Im v2}, b3={Im v1|Im v3}
  float b0 = (lane < 16) ? 1.0f : 0.0f;        // v = e0
  float b1 = 0.f, b2 = 0.f, b3 = 0.f;

  #pragma unroll 4
  for (int k = 0; k < 20; ++k) {
    const float* Ak = Atab + (k << 7);
    v2f Alo, Ahi, Blo, Bhi;
    Alo.x = Ak[lane];      Alo.y = Ak[lane + 32];
    Ahi.x = Ak[lane + 64]; Ahi.y = Ak[lane + 96];
    Blo.x = b0; Blo.y = b1;
    Bhi.x = b2; Bhi.y = b3;
    v8f acc = {0.f,0.f,0.f,0.f,0.f,0.f,0.f,0.f};
    acc = __builtin_amdgcn_wmma_f32_16x16x4_f32(false, Alo, false, Blo, (short)0, acc, false, false);
    acc = __builtin_amdgcn_wmma_f32_16x16x4_f32(false, Ahi, false, Bhi, (short)0, acc, false, false);
    const float w0 = acc[0], w1 = acc[1], w2 = acc[2], w3 = acc[3];

    const float th0 = theta[2*k], th1 = theta[2*k + 1];
    const float a1 = 0.5f * th0 * xv;          // wire 1 angle/2
    const float a2 = 0.5f * th1 * xv;          // wire 2 angle/2
    const float c1 = __cosf(a1), s1 = __sinf(a1);
    const float c2 = __cosf(a2), s2 = __sinf(a2);

    // wire-2 RX: lane-local mix across state VGPRs
    const float n0 = __builtin_fmaf( s2, w3, c2*w0);
    const float n1 = __builtin_fmaf( s2, w2, c2*w1);
    const float n2 = __builtin_fmaf(-s2, w1, c2*w2);
    const float n3 = __builtin_fmaf(-s2, w0, c2*w3);

    // wire-1 RX: mix lane n with lane n^16 (SWAPX16 ds_swizzle, no LDS memory)
    const float sw0 = __int_as_float(__builtin_amdgcn_ds_swizzle(__float_as_int(n0), 0x401f));
    const float sw1 = __int_as_float(__builtin_amdgcn_ds_swizzle(__float_as_int(n1), 0x401f));
    const float sw2 = __int_as_float(__builtin_amdgcn_ds_swizzle(__float_as_int(n2), 0x401f));
    const float sw3 = __int_as_float(__builtin_amdgcn_ds_swizzle(__float_as_int(n3), 0x401f));
    b0 = __builtin_fmaf( s1, sw2, c1*n0);
    b1 = __builtin_fmaf( s1, sw3, c1*n1);
    b2 = __builtin_fmaf(-s1, sw0, c1*n2);
    b3 = __builtin_fmaf(-s1, sw1, c1*n3);
  }

  // final W(phi[20])
  {
    const float* Ak = Atab + (20 << 7);
    v2f Alo, Ahi, Blo, Bhi;
    Alo.x = Ak[lane];      Alo.y = Ak[lane + 32];
    Ahi.x = Ak[lane + 64]; Ahi.y = Ak[lane + 96];
    Blo.x = b0; Blo.y = b1;
    Bhi.x = b2; Bhi.y = b3;
    v8f acc = {0.f,0.f,0.f,0.f,0.f,0.f,0.f,0.f};
    acc = __builtin_amdgcn_wmma_f32_16x16x4_f32(false, Alo, false, Blo, (short)0, acc, false, false);
    acc = __builtin_amdgcn_wmma_f32_16x16x4_f32(false, Ahi, false, Bhi, (short)0, acc, false, false);
    if (lane < 16) out[base + lane] = acc[0];  // Re(U[0,0])
  }
}

extern "C" void kernel_launch(void* const* d_in, const int* in_sizes, int n_in,
                              void* d_out, int out_size, void* d_ws, size_t ws_size,
                              hipStream_t stream) {
  const float* x     = (const float*)d_in[0];   // [N] float32
  const float* theta = (const float*)d_in[1];   // [20,2] float32
  const float* phi   = (const float*)d_in[2];   // [21,6] float32
  float* out  = (float*)d_out;
  float* Atab = (float*)d_ws;                   // 21*128 floats = 10.75 KB

  const int n       = in_sizes[0];
  const int nlayers = in_sizes[2] / 6;          // 21

  dqc1_build_A<<<1, 32, 0, stream>>>(phi, Atab, nlayers);

  const int waves   = (n + 15) / 16;
  const int threads = 256;                      // 8 wave32s per block
  const int blocks  = (waves * 32 + threads - 1) / threads;
  dqc1_main<<<blocks, threads, 0, stream>>>(x, theta, Atab, out, n);
}